// GCN_88502096101880
// MI455X (gfx1250) — compile-verified
//
#include <hip/hip_runtime.h>

#define N_NODESC 50000
#define N_PAD    50048   /* 128 * 391, for full WMMA tiles */
#define N_EDGESC 500000
#define EMB      128
#define NUM_LAYERSC 5
#define VOCABC   128
#define NUM_GRAPHSC 1024
#define NUM_CLASSESC 10
#define BN_EPSF  1e-5f
#define LDW      132     /* LDS row stride (floats) = 128 + TDM pad (4 dwords / 128) */
#define MT       8       /* M tiles (of 16 rows) per block */
#define BM       (16*MT)

typedef float v2f __attribute__((ext_vector_type(2)));
typedef float v4f __attribute__((ext_vector_type(4)));
typedef float v8f __attribute__((ext_vector_type(8)));
typedef unsigned int u32;
typedef u32 u32x4 __attribute__((ext_vector_type(4)));
typedef u32 u32x8 __attribute__((ext_vector_type(8)));

// ---------------------------------------------------------------- utilities
__global__ __launch_bounds__(256) void fill_f32(float* p, float v, int n) {
  int i = blockIdx.x * 256 + threadIdx.x;
  if (i < n) p[i] = v;
}

__global__ __launch_bounds__(256) void degree_kernel(const int* __restrict__ row,
                                                     float* __restrict__ deg) {
  int e = blockIdx.x * 256 + threadIdx.x;
  if (e < N_EDGESC) atomicAdd(&deg[row[e]], 1.0f);
}

__global__ __launch_bounds__(256) void dinv_kernel(const float* __restrict__ deg,
                                                   float* __restrict__ dinv) {
  int i = blockIdx.x * 256 + threadIdx.x;
  if (i < N_NODESC) dinv[i] = rsqrtf(deg[i]);
}

// ------------------------------------------------------------ atom encoder
__global__ __launch_bounds__(128) void atom_encoder(const int* __restrict__ x,
                                                    const float* __restrict__ emb,
                                                    float* __restrict__ h) {
  __shared__ int xi[9];
  int n = blockIdx.x;
  int d = threadIdx.x;
  if (d < 9) xi[d] = x[n * 9 + d];
  __syncthreads();
  float s = 0.f;
#pragma unroll
  for (int i = 0; i < 9; ++i) s += emb[(i * VOCABC + xi[i]) * EMB + d];
  h[n * EMB + d] = s;
}

// ---------------------------------------------------- TDM 2D tile -> LDS
// D# per CDNA5 ISA. 2D tensor: groups 2/3 = NULL.
// pad: every 128 DWORDs (pad_interval code 6) insert 4 DWORDs (pad_amount code 3)
// so a 128-float row lands at LDS stride 132 floats (bank-conflict free).
__device__ __forceinline__ void tdm_load_2d_padded(u32 lds_addr, const void* gptr,
                                                   u32 rowlen, u32 nrows, u32 stride0) {
  unsigned long long ga = (unsigned long long)gptr;
  u32x4 g0;
  g0[0] = 1u;                                            // count=1, user mode
  g0[1] = lds_addr;                                      // LDS byte address
  g0[2] = (u32)ga;                                       // global_addr[31:0]
  g0[3] = (u32)((ga >> 32) & 0x1FFFFFFu) | (2u << 30);   // global_addr[56:32] | type=2
  u32x8 g1;
  g1[0] = (2u << 16)                 // data_size = 4 bytes
        | (1u << 20)                 // pad_enable
        | (6u << 22)                 // pad_interval: 128 DWORDs
        | (3u << 25);                // pad_amount: 4 DWORDs
  g1[1] = (rowlen & 0xFFFFu) << 16;                       // tensor_dim0[15:0]
  g1[2] = ((rowlen >> 16) & 0xFFFFu) | ((nrows & 0xFFFFu) << 16);  // dim0 hi | dim1 lo
  g1[3] = ((nrows >> 16) & 0xFFFFu) | ((rowlen & 0xFFFFu) << 16);  // dim1 hi | tile_dim0
  g1[4] = nrows & 0xFFFFu;                                // tile_dim1 (tile_dim2=0)
  g1[5] = stride0;                                        // tensor_dim0_stride[31:0]
  g1[6] = 0u;                                             // stride0 hi | stride1 lo
  g1[7] = 0u;
  asm volatile("tensor_load_to_lds %0, %1" :: "s"(g0), "s"(g1) : "memory");
}

// ------------------------------------------------- WMMA GEMM: out = h @ W^T
// h: [N_PAD,128], W: [128,128] row-major (W[j][k]); out[n][j] = sum_k h[n][k]*W[j][k]
__global__ __launch_bounds__(256) void gemm_wmma(const float* __restrict__ h,
                                                 const float* __restrict__ W,
                                                 float* __restrict__ out) {
  __shared__ float ws[EMB * LDW];  // W[j][k] at ws[j*LDW+k]  (B = W^T)
  __shared__ float hs[BM * LDW];   // h tile
  const int tid  = threadIdx.x;
  const int row0 = blockIdx.x * BM;

  // Wave 0 issues both TDM DMAs (global -> LDS, no VGPR traffic), waits TENSORcnt.
  if (tid < 32) {
    u32 ws_addr = (u32)(unsigned long long)&ws[0];   // low 32 bits of LDS aperture addr
    u32 hs_addr = (u32)(unsigned long long)&hs[0];
    tdm_load_2d_padded(ws_addr, W, EMB, EMB, EMB);
    tdm_load_2d_padded(hs_addr, h + (long)row0 * EMB, EMB, BM, EMB);
    __builtin_amdgcn_s_wait_tensorcnt(0);
  }
  __syncthreads();

  const int wave = tid >> 5;
  const int lane = tid & 31;
  const int hh   = lane >> 4;   // lane half selects K pair
  const int l15  = lane & 15;
  const int j0   = wave * 16;   // this wave's 16-wide N tile

  v8f acc[MT];
#pragma unroll
  for (int t = 0; t < MT; ++t) acc[t] = {};

#pragma unroll 4
  for (int kk = 0; kk < EMB; kk += 4) {
    const int kb = kk + 2 * hh;                 // A & B: VGPR v, half hh -> K = kk + v + 2*hh
    v2f b;
    b.x = ws[(j0 + l15) * LDW + kb];            // B[kb][j0+l15]   = W[j][kb]
    b.y = ws[(j0 + l15) * LDW + kb + 1];        // B[kb+1][j0+l15]
#pragma unroll
    for (int t = 0; t < MT; ++t) {
      v2f a;
      a.x = hs[(t * 16 + l15) * LDW + kb];      // A[m][kb]
      a.y = hs[(t * 16 + l15) * LDW + kb + 1];  // A[m][kb+1]
      acc[t] = __builtin_amdgcn_wmma_f32_16x16x4_f32(
          false, a, false, b, (short)0, acc[t], false, false);
    }
  }

  // C layout: lane -> N = l15, VGPR r -> M = r + 8*hh
#pragma unroll
  for (int t = 0; t < MT; ++t)
#pragma unroll
    for (int r = 0; r < 8; ++r)
      out[(row0 + t * 16 + r + 8 * hh) * EMB + j0 + l15] = acc[t][r];
}

// --------------------------------------------- edge message + scatter (atomics)
// one edge per wave32; each lane handles 4 consecutive channels
__global__ __launch_bounds__(256) void edge_scatter(const float* __restrict__ hw,
                                                    const int* __restrict__ ei,
                                                    const int* __restrict__ ea,
                                                    const float* __restrict__ bond,
                                                    const float* __restrict__ dinv,
                                                    float* __restrict__ agg) {
  const int wave = threadIdx.x >> 5;
  const int lane = threadIdx.x & 31;
  const int e = blockIdx.x * 8 + wave;
  if (e >= N_EDGESC) return;
  const int r = ei[e];
  const int c = ei[N_EDGESC + e];
  const float norm = dinv[r] * dinv[c];
  const int a0 = ea[e * 3 + 0], a1 = ea[e * 3 + 1], a2 = ea[e * 3 + 2];
  const int d0 = lane * 4;
  v4f hv = *(const v4f*)(hw + (long)r * EMB + d0);
  v4f b0 = *(const v4f*)(bond + (0 * VOCABC + a0) * EMB + d0);
  v4f b1 = *(const v4f*)(bond + (1 * VOCABC + a1) * EMB + d0);
  v4f b2 = *(const v4f*)(bond + (2 * VOCABC + a2) * EMB + d0);
#pragma unroll
  for (int i = 0; i < 4; ++i) {
    float m = hv[i] + b0[i] + b1[i] + b2[i];
    m = m > 0.f ? m : 0.f;
    atomicAdd(&agg[(long)c * EMB + d0 + i], norm * m);
  }
}

// --------------------------- combine: h = agg + relu(hw+root)/deg (+ BN stats)
#define COMB_ROWS 64
__global__ __launch_bounds__(128) void combine_stats(const float* __restrict__ agg,
                                                     const float* __restrict__ hw,
                                                     const float* __restrict__ root,
                                                     const float* __restrict__ deg,
                                                     float* __restrict__ h,
                                                     float* __restrict__ bnsum,
                                                     float* __restrict__ bnsq,
                                                     int doStats) {
  const int d  = threadIdx.x;
  const int n0 = blockIdx.x * COMB_ROWS;
  const float rt = root[d];
  float s = 0.f, sq = 0.f;
  for (int r = 0; r < COMB_ROWS; ++r) {
    int n = n0 + r;
    if (n >= N_NODESC) break;
    float hv = hw[(long)n * EMB + d] + rt;
    hv = hv > 0.f ? hv : 0.f;
    float v = agg[(long)n * EMB + d] + hv / deg[n];
    h[(long)n * EMB + d] = v;
    s += v;
    sq += v * v;
  }
  if (doStats) {
    atomicAdd(&bnsum[d], s);
    atomicAdd(&bnsq[d], sq);
  }
}

__global__ __launch_bounds__(256) void bn_apply(float* __restrict__ h,
                                                const float* __restrict__ bnsum,
                                                const float* __restrict__ bnsq,
                                                const float* __restrict__ gamma,
                                                const float* __restrict__ beta) {
  long idx = (long)blockIdx.x * 256 + threadIdx.x;
  if (idx >= (long)N_NODESC * EMB) return;
  int d = (int)(idx & (EMB - 1));
  float mean = bnsum[d] * (1.0f / N_NODESC);
  float var  = bnsq[d] * (1.0f / N_NODESC) - mean * mean;
  float v = (h[idx] - mean) * rsqrtf(var + BN_EPSF) * gamma[d] + beta[d];
  h[idx] = v > 0.f ? v : 0.f;
}

// ------------------------------------------------------------ pooling + head
__global__ __launch_bounds__(256) void pool_scatter(const float* __restrict__ h,
                                                    const int* __restrict__ batch,
                                                    float* __restrict__ pooled,
                                                    float* __restrict__ counts) {
  const int wave = threadIdx.x >> 5;
  const int lane = threadIdx.x & 31;
  const int n = blockIdx.x * 8 + wave;
  if (n >= N_NODESC) return;
  const int g = batch[n];
  const int d0 = lane * 4;
  v4f v = *(const v4f*)(h + (long)n * EMB + d0);
#pragma unroll
  for (int i = 0; i < 4; ++i) atomicAdd(&pooled[(long)g * EMB + d0 + i], v[i]);
  if (lane == 0) atomicAdd(&counts[g], 1.0f);
}

__global__ __launch_bounds__(256) void project(const float* __restrict__ pooled,
                                               const float* __restrict__ counts,
                                               const float* __restrict__ pw,
                                               float* __restrict__ out) {
  int idx = blockIdx.x * 256 + threadIdx.x;
  if (idx >= NUM_GRAPHSC * NUM_CLASSESC) return;
  int g = idx / NUM_CLASSESC, c = idx % NUM_CLASSESC;
  float cnt = counts[g];
  cnt = cnt > 1.f ? cnt : 1.f;
  float s = 0.f;
#pragma unroll 4
  for (int d = 0; d < EMB; ++d) s += pooled[g * EMB + d] * pw[c * EMB + d];
  out[idx] = s / cnt;
}

// ---------------------------------------------------------------- launcher
extern "C" void kernel_launch(void* const* d_in, const int* in_sizes, int n_in,
                              void* d_out, int out_size, void* d_ws, size_t ws_size,
                              hipStream_t stream) {
  const int*   x        = (const int*)d_in[0];
  const int*   ei       = (const int*)d_in[1];
  const int*   ea       = (const int*)d_in[2];
  const int*   batch    = (const int*)d_in[3];
  const float* atom_emb = (const float*)d_in[4];
  const float* bond_emb = (const float*)d_in[5];
  const float* fc_w     = (const float*)d_in[6];
  const float* root_emb = (const float*)d_in[7];
  const float* bn_gamma = (const float*)d_in[8];
  const float* bn_beta  = (const float*)d_in[9];
  const float* pred_w   = (const float*)d_in[10];
  float* out = (float*)d_out;

  float* wsf    = (float*)d_ws;
  float* h      = wsf;                         // N_PAD*128
  float* hw     = h + (long)N_PAD * EMB;       // N_PAD*128
  float* agg    = hw + (long)N_PAD * EMB;      // N*128
  float* deg    = agg + (long)N_NODESC * EMB;  // N
  float* dinv   = deg + N_NODESC;              // N
  float* bnsum  = dinv + N_NODESC;             // 128
  float* bnsq   = bnsum + EMB;                 // 128 (contiguous with bnsum)
  float* pooled = bnsq + EMB;                  // 1024*128
  float* counts = pooled + (long)NUM_GRAPHSC * EMB;  // 1024 (contiguous)

  (void)in_sizes; (void)n_in; (void)out_size; (void)ws_size;

  // degrees (reused across layers)
  fill_f32<<<(N_NODESC + 255) / 256, 256, 0, stream>>>(deg, 1.0f, N_NODESC);
  degree_kernel<<<(N_EDGESC + 255) / 256, 256, 0, stream>>>(ei, deg);
  dinv_kernel<<<(N_NODESC + 255) / 256, 256, 0, stream>>>(deg, dinv);

  // atom encoder
  atom_encoder<<<N_NODESC, 128, 0, stream>>>(x, atom_emb, h);

  for (int L = 0; L < NUM_LAYERSC; ++L) {
    gemm_wmma<<<N_PAD / BM, 256, 0, stream>>>(h, fc_w + (long)L * EMB * EMB, hw);
    fill_f32<<<((long)N_NODESC * EMB + 255) / 256, 256, 0, stream>>>(agg, 0.f,
                                                                     N_NODESC * EMB);
    edge_scatter<<<N_EDGESC / 8, 256, 0, stream>>>(
        hw, ei, ea, bond_emb + (long)L * 3 * VOCABC * EMB, dinv, agg);
    fill_f32<<<1, 256, 0, stream>>>(bnsum, 0.f, 2 * EMB);  // bnsum + bnsq
    combine_stats<<<(N_NODESC + COMB_ROWS - 1) / COMB_ROWS, 128, 0, stream>>>(
        agg, hw, root_emb + L * EMB, deg, h, bnsum, bnsq,
        (L < NUM_LAYERSC - 1) ? 1 : 0);
    if (L < NUM_LAYERSC - 1)
      bn_apply<<<((long)N_NODESC * EMB + 255) / 256, 256, 0, stream>>>(
          h, bnsum, bnsq, bn_gamma + L * EMB, bn_beta + L * EMB);
  }

  fill_f32<<<(NUM_GRAPHSC * EMB + NUM_GRAPHSC + 255) / 256, 256, 0, stream>>>(
      pooled, 0.f, NUM_GRAPHSC * EMB + NUM_GRAPHSC);  // pooled + counts
  pool_scatter<<<N_NODESC / 8, 256, 0, stream>>>(h, batch, pooled, counts);
  project<<<(NUM_GRAPHSC * NUM_CLASSESC + 255) / 256, 256, 0, stream>>>(
      pooled, counts, pred_w, out);
}